// AllocatingLayer_1580547971241
// MI455X (gfx1250) — compile-verified
//
#include <hip/hip_runtime.h>
#include <stdint.h>

typedef float v2f __attribute__((ext_vector_type(2)));
typedef float v8f __attribute__((ext_vector_type(8)));

#define KDIM 2048
#define W_RESOURCE 100.0f

// One block = 8 waves; each wave computes a 16-row output tile of one batch row
// via v_wmma_f32_16x16x4_f32 accumulation over K. grid = (B, 16).
__global__ __launch_bounds__(256)
void allocating_layer_kernel(const float* __restrict__ values,
                             const float* __restrict__ weights,
                             float* __restrict__ out)
{
    __shared__ __align__(16) float vals[KDIM];
    __shared__ __align__(16) float wts[KDIM];

    const uint32_t tid  = threadIdx.x;
    const int      b    = blockIdx.x;

    // ---- Stage this batch row into LDS with CDNA5 async global->LDS DMA ----
    // 256 threads x 16B = 4096B per instruction; offset:4096 advances BOTH the
    // global and LDS addresses (CDNA5 ISA GVS pseudocode).
    {
        const uint32_t byte_off = tid * 16u;                         // 0..4080
        uint32_t lds_v = (uint32_t)(uintptr_t)(&vals[0]) + byte_off; // LDS byte addr
        uint32_t lds_w = (uint32_t)(uintptr_t)(&wts[0])  + byte_off;
        const float* vrow = values  + (size_t)b * KDIM;
        const float* wrow = weights + (size_t)b * KDIM;
        asm volatile(
            "global_load_async_to_lds_b128 %0, %1, %2 offset:0\n\t"
            "global_load_async_to_lds_b128 %0, %1, %2 offset:4096"
            :
            : "v"(lds_v), "v"(byte_off), "s"(vrow)
            : "memory");
        asm volatile(
            "global_load_async_to_lds_b128 %0, %1, %2 offset:0\n\t"
            "global_load_async_to_lds_b128 %0, %1, %2 offset:4096"
            :
            : "v"(lds_w), "v"(byte_off), "s"(wrow)
            : "memory");
        asm volatile("s_wait_asynccnt 0" ::: "memory");
    }
    __syncthreads();

    // ---- Per-wave 16-row tile ----
    const int lane = tid & 31;
    const int wave = tid >> 5;
    const int i0   = (blockIdx.y * 8 + wave) * 16;   // tile base row index
    const int m    = lane & 15;                      // A-matrix row for this lane
    const int hi   = lane >> 4;                      // lane half: K pair selector

    const float vi = vals[i0 + m];                   // threshold value for row m

    // A 16x4 f32 layout: lane holds K = {2*hi, 2*hi+1} (VGPR v -> K = v + 2*hi).
    // B 4x16 f32 with w replicated across all 16 N columns -> lane supplies the
    // same K pair of weights; every D column then equals load[] for this tile.
    v8f acc = {};
    #pragma unroll 8
    for (int k0 = 0; k0 < KDIM; k0 += 4) {
        const int kk = k0 + 2 * hi;
        const float2 vv = *(const float2*)&vals[kk];
        const float2 ww = *(const float2*)&wts[kk];
        v2f a, bb;
        a.x  = (vv.x >= vi) ? 1.0f : 0.0f;           // better[m, kk]
        a.y  = (vv.y >= vi) ? 1.0f : 0.0f;           // better[m, kk+1]
        bb.x = ww.x;
        bb.y = ww.y;
        acc = __builtin_amdgcn_wmma_f32_16x16x4_f32(
            /*neg_a=*/false, a, /*neg_b=*/false, bb,
            /*c_mod=*/(short)0, acc, /*reuse_a=*/false, /*reuse_b=*/false);
    }

    // D layout (32-bit C/D 16x16): VGPR r holds M=r (lanes 0-15) or M=r+8
    // (lanes 16-31); columns are identical since B was N-replicated.
    if ((lane & 15) == 0) {
        const int row0 = i0 + hi * 8;
        float* orow = out + (size_t)b * KDIM + row0;
        #pragma unroll
        for (int r = 0; r < 8; ++r) {
            const float w = wts[row0 + r];
            orow[r] = (W_RESOURCE - acc[r] >= 0.0f) ? w : 0.0f;
        }
    }
}

extern "C" void kernel_launch(void* const* d_in, const int* in_sizes, int n_in,
                              void* d_out, int out_size, void* d_ws, size_t ws_size,
                              hipStream_t stream) {
    const float* values  = (const float*)d_in[0];
    const float* weights = (const float*)d_in[1];
    float*       out     = (float*)d_out;
    const int B = in_sizes[0] / KDIM;                // 32 for the reference shapes
    dim3 grid(B, 16);                                // 16 blocks/row x 8 waves x 16 rows = 2048
    allocating_layer_kernel<<<grid, 256, 0, stream>>>(values, weights, out);
}